// G1_sub1_update_84937273245885
// MI455X (gfx1250) — compile-verified
//
#include <hip/hip_runtime.h>

typedef float v2f __attribute__((ext_vector_type(2)));
typedef float v4f __attribute__((ext_vector_type(4)));
typedef float v8f __attribute__((ext_vector_type(8)));

#define DCOLS   128
#define NTILES  (DCOLS / 16)   // 8 column tiles of 16
#define CHUNKS  512            // waves per column tile (row-chunk parallelism)
#define B_BLOCKS 512           // 512 blocks * 8 waves = 4096 waves = 8 tiles * 512 chunks

// ---------------------------------------------------------------------------
// Kernel B: column sums via V_WMMA_F32_16X16X4_F32, 4x unrolled.
// A = all-ones 16x4  =>  D[m,n] += sum_k B[k,n]  (replicated over m).
// Each wave owns (tile, chunk): 16 columns, 16 rows per iteration.
// 8 b32 loads in flight per iteration feed 4 chained WMMAs.
// Sum is invariant to the K-permutation inside B's VGPR layout; lanes 0-15
// carry N=0..15 (matches documented C/D layout).
// ---------------------------------------------------------------------------
__global__ void colsum_wmma_kernel(const float* __restrict__ emb,
                                   float* __restrict__ partials,
                                   int M) {
    const int lane = threadIdx.x & 31;             // wave32
    const int wv   = threadIdx.x >> 5;
    const int wave_global = blockIdx.x * (blockDim.x >> 5) + wv;
    const int tile  = wave_global & (NTILES - 1);
    // chunk is wave-uniform: make it scalar so loop control stays SALU and
    // EXEC remains all-ones around the WMMAs.
    const int chunk = __builtin_amdgcn_readfirstlane(wave_global >> 3);
    const int col   = tile * 16 + (lane & 15);
    const int half  = lane >> 4;                   // K-row parity within group

    v8f c = {0.f, 0.f, 0.f, 0.f, 0.f, 0.f, 0.f, 0.f};
    v2f a; a[0] = 1.0f; a[1] = 1.0f;               // all-ones A (16x4)

    const int stride = CHUNKS * 16;                // 8192 rows / iteration ring
    int r = chunk * 16;
    for (; r + 15 < M; r += stride) {
        const float* base = emb + (size_t)r * DCOLS + col;
        v2f b0, b1, b2, b3;
        b0[0] = base[(size_t)(half)      * DCOLS];
        b0[1] = base[(size_t)(half + 2)  * DCOLS];
        b1[0] = base[(size_t)(half + 4)  * DCOLS];
        b1[1] = base[(size_t)(half + 6)  * DCOLS];
        b2[0] = base[(size_t)(half + 8)  * DCOLS];
        b2[1] = base[(size_t)(half + 10) * DCOLS];
        b3[0] = base[(size_t)(half + 12) * DCOLS];
        b3[1] = base[(size_t)(half + 14) * DCOLS];
        // (neg_a, A, neg_b, B, c_mod, C, reuse_a, reuse_b)
        c = __builtin_amdgcn_wmma_f32_16x16x4_f32(false, a, false, b0, (short)0, c, false, false);
        c = __builtin_amdgcn_wmma_f32_16x16x4_f32(false, a, false, b1, (short)0, c, false, false);
        c = __builtin_amdgcn_wmma_f32_16x16x4_f32(false, a, false, b2, (short)0, c, false, false);
        c = __builtin_amdgcn_wmma_f32_16x16x4_f32(false, a, false, b3, (short)0, c, false, false);
    }

    // Tail rows (M % 16 != 0): plain VALU adds, only chunk 0 covers them.
    float extra = 0.f;
    if (chunk == 0) {
        for (int t = M & ~15; t < M; ++t)
            extra += emb[(size_t)t * DCOLS + col];
    }

    // c[0] on lanes 0-15 holds (M=0, N=lane) = column-sum partial.
    if (lane < 16)
        partials[(size_t)chunk * DCOLS + col] = c[0] + extra;
}

// ---------------------------------------------------------------------------
// Kernel C: deterministic fixed-order reduction of per-chunk partials.
// ---------------------------------------------------------------------------
__global__ void reduce_partials_kernel(const float* __restrict__ partials,
                                       float* __restrict__ total) {
    const int col = threadIdx.x;                   // 128 threads
    float s = 0.f;
    for (int ck = 0; ck < CHUNKS; ++ck)
        s += partials[(size_t)ck * DCOLS + col];
    total[col] = s;
}

// ---------------------------------------------------------------------------
// Kernel D: degree histogram. Integer atomics => exact & deterministic.
// adj_rows is streamed once -> non-temporal loads (don't evict emb from L2).
// ---------------------------------------------------------------------------
__global__ void degree_kernel(const int* __restrict__ rows,
                              int* __restrict__ deg, int E) {
    int i      = blockIdx.x * blockDim.x + threadIdx.x;
    int stride = gridDim.x * blockDim.x;
    for (; i < E; i += stride) {
        int ri = __builtin_nontemporal_load(rows + i);
        atomicAdd(&deg[ri], 1);
    }
}

// ---------------------------------------------------------------------------
// Kernel E: out[i,d] = (emb[i,d] + total[d]) * (1 - total[d]/(1+deg[i]))
// One wave per row (32 lanes x float4). Rows processed in REVERSE order so
// the emb tail still resident in the 192MB L2 (from the forward-streaming
// column-sum pass) is hit first. emb read is last-use and the out store is
// write-once -> both non-temporal to preserve L2 residency.
// ---------------------------------------------------------------------------
__global__ void finalize_kernel(const float* __restrict__ emb,
                                const int*   __restrict__ deg,
                                const float* __restrict__ total,
                                float*       __restrict__ out, int M) {
    __shared__ __align__(16) float tot[DCOLS];
    if (threadIdx.x < DCOLS) tot[threadIdx.x] = total[threadIdx.x];
    __syncthreads();

    const int    cq      = threadIdx.x & 31;                    // float4 slot
    const size_t row_idx = (size_t)blockIdx.x * 8 + (threadIdx.x >> 5);
    if (row_idx >= (size_t)M) return;
    const size_t row = (size_t)M - 1 - row_idx;                 // reversed

    const size_t q  = row * 32 + cq;
    const v4f e  = __builtin_nontemporal_load(
                       reinterpret_cast<const v4f*>(emb) + q);  // last use
    const v4f t4 = reinterpret_cast<const v4f*>(tot)[cq];
    const float inv = 1.0f / (1.0f + (float)deg[row]);

    v4f res;
    res[0] = (e[0] + t4[0]) * (1.0f - t4[0] * inv);
    res[1] = (e[1] + t4[1]) * (1.0f - t4[1] * inv);
    res[2] = (e[2] + t4[2]) * (1.0f - t4[2] * inv);
    res[3] = (e[3] + t4[3]) * (1.0f - t4[3] * inv);
    __builtin_nontemporal_store(res, reinterpret_cast<v4f*>(out) + q);
}

// ---------------------------------------------------------------------------
extern "C" void kernel_launch(void* const* d_in, const int* in_sizes, int n_in,
                              void* d_out, int out_size, void* d_ws, size_t ws_size,
                              hipStream_t stream) {
    const float* emb  = (const float*)d_in[0];   // (M, 128) f32
    const int*   rows = (const int*)d_in[1];     // (E,) i32
    float*       out  = (float*)d_out;           // (M, 128) f32

    const int M = in_sizes[0] / DCOLS;           // 500000
    const int E = in_sizes[1];                   // 8000000

    // Workspace layout: [total 128f][partials CHUNKS*128f][deg M*i32]
    float* total    = (float*)d_ws;
    float* partials = total + DCOLS;
    int*   deg      = (int*)(partials + (size_t)CHUNKS * DCOLS);

    hipMemsetAsync(deg, 0, (size_t)M * sizeof(int), stream);

    colsum_wmma_kernel<<<B_BLOCKS, 256, 0, stream>>>(emb, partials, M);
    reduce_partials_kernel<<<1, DCOLS, 0, stream>>>(partials, total);
    degree_kernel<<<2048, 256, 0, stream>>>(rows, deg, E);
    finalize_kernel<<<(M + 7) / 8, 256, 0, stream>>>(emb, deg, total, out, M);
}